// Neuron_33500744909351
// MI455X (gfx1250) — compile-verified
//
#include <hip/hip_runtime.h>
#include <math.h>

typedef float v2f __attribute__((ext_vector_type(2)));
typedef float v8f __attribute__((ext_vector_type(8)));

#define INPUT_SIZE   4096
#define CONTEXT_SIZE 128
#define BATCH        8192
#define LR           0.01f
#define WEIGHT_CLIP  5.0f
// logit(0.01) = log(0.01) - log1p(-0.01)
#define OUT_LO (-4.595119850134589f)
#define OUT_HI ( 4.595119850134589f)

#define BLOCK   512            // 16 waves per block
#define NWAVES  16

// ---------------- Kernel A: context hashing -> ctx_idx[b] ----------------
__global__ void ctx_kernel(const float* __restrict__ cin,
                           const float* __restrict__ cmap,
                           const float* __restrict__ cbias,
                           int* __restrict__ ctx) {
  int b = blockIdx.x * blockDim.x + threadIdx.x;
  if (b >= BATCH) return;
  float d0 = 0.f, d1 = 0.f, d2 = 0.f, d3 = 0.f;
  #pragma unroll 4
  for (int j = 0; j < CONTEXT_SIZE; ++j) {
    float x = cin[j * BATCH + b];               // coalesced along b
    d0 = fmaf(cmap[0 * CONTEXT_SIZE + j], x, d0);
    d1 = fmaf(cmap[1 * CONTEXT_SIZE + j], x, d1);
    d2 = fmaf(cmap[2 * CONTEXT_SIZE + j], x, d2);
    d3 = fmaf(cmap[3 * CONTEXT_SIZE + j], x, d3);
  }
  int idx = (int)(d0 > cbias[0])
          | ((int)(d1 > cbias[1]) << 1)
          | ((int)(d2 > cbias[2]) << 2)
          | ((int)(d3 > cbias[3]) << 3);
  ctx[b] = idx;
}

// ---------------- Kernel B: out = diag-select(W @ logits), coef, M -------
// One block per 16-column tile of logits; 16 waves split K = INPUT_SIZE,
// each accumulating a partial 16x16 tile with V_WMMA_F32_16X16X4_F32.
// Fixed-order LDS reduction keeps the result deterministic.
__global__ void __launch_bounds__(BLOCK)
fwd_kernel(const float* __restrict__ logits,
           const float* __restrict__ W,
           const float* __restrict__ targets,
           const int* __restrict__ ctx,
           float* __restrict__ out,
           float* __restrict__ Mmat) {
  __shared__ float part[NWAVES * 256];     // 16 waves x (8 regs x 32 lanes) = 16 KB
  const int wave = threadIdx.x >> 5;
  const int lane = threadIdx.x & 31;
  const int b0   = blockIdx.x << 4;                         // 0..511 tiles
  const int row  = lane & 15;
  const int kh   = (lane >> 4) << 1;                        // 0 or 2
  const int k0   = wave * (INPUT_SIZE / NWAVES);            // 256-wide K slice

  v8f acc = {0.f, 0.f, 0.f, 0.f, 0.f, 0.f, 0.f, 0.f};
  const float* wrow = W + (size_t)row * INPUT_SIZE + kh;             // A side
  const float* lcol = logits + (size_t)kh * BATCH + b0 + row;        // B side

  #pragma unroll 4
  for (int k = k0; k < k0 + INPUT_SIZE / NWAVES; k += 4) {
    v2f a = *(const v2f*)(wrow + k);                 // W[row][k+kh], W[row][k+kh+1]
    v2f bb;
    bb.x = lcol[(size_t)k * BATCH];                  // logits[k+kh][b0+row]
    bb.y = lcol[(size_t)k * BATCH + BATCH];          // logits[k+kh+1][b0+row]
    acc = __builtin_amdgcn_wmma_f32_16x16x4_f32(
        false, a, false, bb, (short)0, acc, false, false);
  }

  #pragma unroll
  for (int r = 0; r < 8; ++r)
    part[wave * 256 + r * 32 + lane] = acc[r];       // lane-fastest: no bank conflicts
  __syncthreads();

  if (wave == 0) {
    float tile[8];
    #pragma unroll
    for (int r = 0; r < 8; ++r) {                    // fixed-order sum -> deterministic
      float s = part[r * 32 + lane];
      #pragma unroll
      for (int w2 = 1; w2 < NWAVES; ++w2)
        s += part[w2 * 256 + r * 32 + lane];
      tile[r] = s;
    }

    // Diagonal select: column n needs P[ctx[b0+n]][n].
    const int n = lane & 15;
    const int c = ctx[b0 + n];
    const int c7 = c & 7;
    float sel = tile[0];
    sel = (c7 == 1) ? tile[1] : sel;
    sel = (c7 == 2) ? tile[2] : sel;
    sel = (c7 == 3) ? tile[3] : sel;
    sel = (c7 == 4) ? tile[4] : sel;
    sel = (c7 == 5) ? tile[5] : sel;
    sel = (c7 == 6) ? tile[6] : sel;
    sel = (c7 == 7) ? tile[7] : sel;
    const int src = n + ((c >> 3) << 4);             // lane holding M = c
    float val = __shfl(sel, src, 32);

    float o = fminf(fmaxf(val, OUT_LO), OUT_HI);
    if (lane < 16) {
      out[b0 + n] = o;
      float t    = targets[b0 + n];
      float sig  = 1.f / (1.f + expf(-o));
      float coef = LR * (sig - t);
      #pragma unroll
      for (int cc = 0; cc < 16; ++cc)
        Mmat[(size_t)(b0 + n) * 16 + cc] = (cc == c) ? coef : 0.f;
    }
  }
}

// ---------------- Kernel C: new_W = clip(W - logits @ M) -----------------
// One block per 16-row tile of i; 16 waves split K = BATCH, partials
// reduced in fixed order, then wave 0 applies the clipped update.
__global__ void __launch_bounds__(BLOCK)
upd_kernel(const float* __restrict__ logits,
           const float* __restrict__ W,
           const float* __restrict__ Mmat,
           float* __restrict__ Wout) {
  __shared__ float part[NWAVES * 256];
  const int wave = threadIdx.x >> 5;
  const int lane = threadIdx.x & 31;
  const int i0   = blockIdx.x << 4;                         // 0..255 tiles
  const int row  = lane & 15;
  const int kh   = (lane >> 4) << 1;                        // 0 or 2
  const int k0   = wave * (BATCH / NWAVES);                 // 512-wide K slice

  v8f acc = {0.f, 0.f, 0.f, 0.f, 0.f, 0.f, 0.f, 0.f};
  const float* arow = logits + (size_t)(i0 + row) * BATCH + kh;   // A side
  const float* brow = Mmat + (size_t)kh * 16 + row;               // B side

  #pragma unroll 4
  for (int k = k0; k < k0 + BATCH / NWAVES; k += 4) {
    v2f a = *(const v2f*)(arow + k);                 // logits[i0+row][k+kh .. +1]
    v2f bb;
    bb.x = brow[(size_t)k * 16];                     // M[k+kh][row]
    bb.y = brow[(size_t)k * 16 + 16];                // M[k+kh+1][row]
    acc = __builtin_amdgcn_wmma_f32_16x16x4_f32(
        false, a, false, bb, (short)0, acc, false, false);
  }

  #pragma unroll
  for (int r = 0; r < 8; ++r)
    part[wave * 256 + r * 32 + lane] = acc[r];
  __syncthreads();

  if (wave == 0) {
    // D layout: reg r -> (i = i0 + r + 8*(lane>>4), c = lane&15)
    const int c  = lane & 15;
    const int ib = i0 + ((lane >> 4) << 3);
    #pragma unroll
    for (int r = 0; r < 8; ++r) {
      float d = part[r * 32 + lane];
      #pragma unroll
      for (int w2 = 1; w2 < NWAVES; ++w2)
        d += part[w2 * 256 + r * 32 + lane];         // fixed order -> deterministic
      int i = ib + r;
      float wold = W[(size_t)c * INPUT_SIZE + i];
      float nw = wold - d;
      nw = fminf(fmaxf(nw, -WEIGHT_CLIP), WEIGHT_CLIP);
      Wout[(size_t)c * INPUT_SIZE + i] = nw;
    }
  }
}

extern "C" void kernel_launch(void* const* d_in, const int* in_sizes, int n_in,
                              void* d_out, int out_size, void* d_ws, size_t ws_size,
                              hipStream_t stream) {
  const float* logits  = (const float*)d_in[0];   // (4096, 8192)
  const float* cin     = (const float*)d_in[1];   // (128, 8192)
  const float* targets = (const float*)d_in[2];   // (8192,)
  const float* cmaps   = (const float*)d_in[3];   // (4, 128)
  const float* cbias   = (const float*)d_in[4];   // (4, 1)
  const float* weights = (const float*)d_in[5];   // (16, 4096)

  float* out  = (float*)d_out;            // first 8192 floats
  float* Wout = out + BATCH;              // next 16*4096 floats

  char* ws = (char*)d_ws;
  int*   ctx  = (int*)ws;                            // 8192 ints  (32 KB)
  float* Mmat = (float*)(ws + BATCH * sizeof(int));  // 8192*16 f  (512 KB)

  ctx_kernel<<<BATCH / 256, 256, 0, stream>>>(cin, cmaps, cbias, ctx);

  // 512 b-tiles, one block (16 waves, K-split) each -> 8192 waves
  fwd_kernel<<<BATCH / 16, BLOCK, 0, stream>>>(logits, weights, targets,
                                               ctx, out, Mmat);

  // 256 i-tiles, one block (16 waves, K-split) each -> 4096 waves
  upd_kernel<<<INPUT_SIZE / 16, BLOCK, 0, stream>>>(logits, weights,
                                                    Mmat, Wout);
}